// NormD_26860725469803
// MI455X (gfx1250) — compile-verified
//
#include <hip/hip_runtime.h>
#include <stdint.h>

#define NSLOTS 24
#define BB 4
#define CC 32
#define NN 200
#define TT 2048
#define EPSF 1e-5f

typedef float v4f __attribute__((ext_vector_type(4)));

// d_ws layout in floats:
//   [0,768)     per-slot-per-channel sum
//   [768,1536)  per-slot-per-channel sum of squares
//   [1536,1560) per-slot counts
//   [1600,2368) mean table
//   [2368,3136) rstd table
#define WS_SUM  0
#define WS_SQ   768
#define WS_CNT  1536
#define WS_MEAN 1600
#define WS_RSTD 2368

__global__ void nd_zero(float* __restrict__ ws) {
  int i = blockIdx.x * blockDim.x + threadIdx.x;
  if (i < WS_MEAN) ws[i] = 0.0f;
}

// time is int64 with values in [0,24): read the low 32-bit word (little endian).
__global__ void nd_count(const int* __restrict__ time2, float* __restrict__ cnt) {
  int i = blockIdx.x * blockDim.x + threadIdx.x;   // 0 .. B*T-1
  int slot = time2[2 * i];
  atomicAdd(cnt + slot, 1.0f);
}

__global__ __launch_bounds__(256)
void nd_stats(const float* __restrict__ x, const int* __restrict__ time2,
              float* __restrict__ gsum, float* __restrict__ gsq) {
  __shared__ float lsum[NSLOTS * CC];
  __shared__ float lsq [NSLOTS * CC];
  const int tid = threadIdx.x;
  for (int i = tid; i < NSLOTS * CC; i += 256) { lsum[i] = 0.f; lsq[i] = 0.f; }
  __syncthreads();

  const int b  = blockIdx.z;             // batch
  const int t  = blockIdx.y * 256 + tid; // one t per thread -> coalesced 128B/wave
  const int n0 = blockIdx.x * 8;         // 25 n-chunks of 8
  const int slot = time2[2 * (b * TT + t)];

  const float* xb = x + ((size_t)(b * CC) * NN + n0) * TT + t;
  for (int c = 0; c < CC; ++c) {
    const float* xc = xb + (size_t)c * NN * TT;
    float a0 = 0.f, a1 = 0.f;
#pragma unroll
    for (int nn = 0; nn < 8; ++nn) {
      float v = xc[(size_t)nn * TT];     // RT load: deliberately fills L2 for pass 3
      a0 += v;
      a1 += v * v;
    }
    atomicAdd(&lsum[slot * CC + c], a0); // ds_add_f32
    atomicAdd(&lsq [slot * CC + c], a1);
  }
  __syncthreads();
  for (int i = tid; i < NSLOTS * CC; i += 256) {
    atomicAdd(gsum + i, lsum[i]);        // global_atomic_add_f32
    atomicAdd(gsq  + i, lsq[i]);
  }
}

__global__ void nd_finalize(const float* __restrict__ gsum, const float* __restrict__ gsq,
                            const float* __restrict__ gcnt, float* __restrict__ mean,
                            float* __restrict__ rstd) {
  int i = threadIdx.x;                   // 768 = 24*32 threads
  int slot = i / CC;
  float cnt = fmaxf(gcnt[slot], 1e-12f); // matches reference denom clamp
  float inv = 1.0f / (cnt * (float)NN);  // extra 1/N from mean over n
  float m  = gsum[i] * inv;
  float m2 = gsq[i]  * inv;
  mean[i] = m;
  rstd[i] = 1.0f / sqrtf(m2 - m * m + EPSF);
}

__global__ __launch_bounds__(256)
void nd_norm(const float* __restrict__ x, const int* __restrict__ time2,
             const float* __restrict__ mean, const float* __restrict__ rstd,
             float* __restrict__ out) {
  __shared__ float lx[TT];               // one 8KB row staged via async copy
  __shared__ float sm[NSLOTS];
  __shared__ float sr[NSLOTS];
  const int tid = threadIdx.x;
  // Reverse traversal: pass 1 streamed x forward, so L2 (192MB of 210MB) holds
  // the tail of x. Starting from the last row turns most pass-3 reads into L2 hits.
  const int row = (BB * CC * NN - 1) - (int)blockIdx.x;
  const int b = row / (CC * NN);
  const int c = (row / NN) % CC;

  if (tid < NSLOTS) {
    sm[tid] = mean[tid * CC + c];
    sr[tid] = rstd[tid * CC + c];
  }

  const v4f* __restrict__ xr = (const v4f*)(x + (size_t)row * TT);
  v4f* __restrict__ orow     = (v4f*)(out + (size_t)row * TT);

  // Generic pointers into LDS carry the LDS byte offset in their low 32 bits
  // (aperture form, ISA 10.2); that offset is what VDST of the async op wants.
  uint32_t lbase = (uint32_t)(uintptr_t)&lx[0];

  // Stage the row: 512 lanes-worth of b128 async global->LDS copies (ASYNCcnt).
  // th:TH_LOAD_LU = last-use: each x line is read exactly once here, discard on
  // hit so the reversed sweep keeps freeing L2 capacity ahead of itself.
#pragma unroll
  for (int k = 0; k < 2; ++k) {
    int i = tid + k * 256;               // float4 index in row, 0..511
    asm volatile("global_load_async_to_lds_b128 %0, %1, off th:TH_LOAD_LU"
                 :
                 : "v"(lbase + (uint32_t)i * 16u), "v"(xr + i)
                 : "memory");
  }
  asm volatile("s_wait_asynccnt 0" ::: "memory");
  __syncthreads();                       // cross-wave visibility of LDS + sm/sr

  const int tbase = b * TT;
#pragma unroll
  for (int k = 0; k < 2; ++k) {
    int i = tid + k * 256;
    v4f v = ((const v4f*)lx)[i];
    int t0 = i * 4;
    int s0 = time2[2 * (tbase + t0 + 0)];
    int s1 = time2[2 * (tbase + t0 + 1)];
    int s2 = time2[2 * (tbase + t0 + 2)];
    int s3 = time2[2 * (tbase + t0 + 3)];
    v4f o;
    o.x = (v.x - sm[s0]) * sr[s0];
    o.y = (v.y - sm[s1]) * sr[s1];
    o.z = (v.z - sm[s2]) * sr[s2];
    o.w = (v.w - sm[s3]) * sr[s3];
    // NT store: don't let 210MB of output evict the x lines we re-read from L2.
    __builtin_nontemporal_store(o, orow + i);
  }
}

extern "C" void kernel_launch(void* const* d_in, const int* in_sizes, int n_in,
                              void* d_out, int out_size, void* d_ws, size_t ws_size,
                              hipStream_t stream) {
  (void)in_sizes; (void)n_in; (void)out_size; (void)ws_size;
  const float* x     = (const float*)d_in[0];
  const int*   time2 = (const int*)d_in[1];   // int64 viewed as int pairs
  float* out = (float*)d_out;
  float* ws  = (float*)d_ws;

  float* gsum = ws + WS_SUM;
  float* gsq  = ws + WS_SQ;
  float* gcnt = ws + WS_CNT;
  float* mean = ws + WS_MEAN;
  float* rstd = ws + WS_RSTD;

  nd_zero    <<<(WS_MEAN + 255) / 256, 256, 0, stream>>>(ws);
  nd_count   <<<(BB * TT) / 256, 256, 0, stream>>>(time2, gcnt);
  nd_stats   <<<dim3(NN / 8, TT / 256, BB), 256, 0, stream>>>(x, time2, gsum, gsq);
  nd_finalize<<<1, NSLOTS * CC, 0, stream>>>(gsum, gsq, gcnt, mean, rstd);
  nd_norm    <<<BB * CC * NN, 256, 0, stream>>>(x, time2, mean, rstd, out);
}